// _ProposalLayer_81071802679971
// MI455X (gfx1250) — compile-verified
//
#include <hip/hip_runtime.h>
#include <stdint.h>

typedef float v2f __attribute__((ext_vector_type(2)));
typedef float v8f __attribute__((ext_vector_type(8)));

#define BB_    8
#define AA_    9
#define HH_    64
#define WW_    64
#define NN_    (AA_*HH_*WW_)     // 36864 proposals per batch
#define NPAD_  65536             // padded pow2 for bitonic sort
#define PRE_   6000
#define POST_  300
#define THRESH_ 0.7f
#define NT_    (PRE_/16)         // 375 16-wide tiles (6000 is exactly 375*16)
#define CHUNK_ 4096              // LDS-resident sort chunk (32 KB of u64 keys)
#define LTH_   1024              // threads per sort block

static_assert(PRE_ % 16 == 0, "tile");
static_assert(NPAD_ % CHUNK_ == 0, "chunk");

// ---- workspace layout (bytes), all 256-aligned, total ~45.9 MB (L2-resident) ----
#define OFF_KEYS 0u                         // B*65536 u64          = 4,194,304
#define OFF_PROP 4194304u                   // B*N*4 f32            = 4,718,592
#define OFF_BOX  8913152u                   // B*6000*4 f32         =   768,000
#define OFF_AREA 9681408u                   // B*6000 f32           =   192,000
#define OFF_MASK 9873664u                   // B*6000*375 u16       = 36,000,000

// ---------------- 1) decode boxes + build sort keys ----------------
__global__ void decode_kernel(const float* __restrict__ scores,
                              const float* __restrict__ deltas,
                              const float* __restrict__ anchors,
                              const int* __restrict__ imgw,
                              const int* __restrict__ imgh,
                              float* __restrict__ proposals,
                              unsigned long long* __restrict__ keys) {
  int t = blockIdx.x * blockDim.x + threadIdx.x;
  if (t >= BB_ * NPAD_) return;
  int b = t >> 16;
  int n = t & (NPAD_ - 1);
  if (n >= NN_) { keys[t] = ~0ULL; return; }   // sentinel sorts last
  int a  = n % AA_;
  int hw = n / AA_;
  int w  = hw % WW_;
  int h  = hw / WW_;
  // fg score: scores[b, A + a, h, w]
  float s = scores[(((size_t)b * (2*AA_) + AA_ + a) * HH_ + h) * WW_ + w];
  // monotonic key: larger score -> smaller key; ties broken by smaller index
  unsigned u = __float_as_uint(s);
  u = (u & 0x80000000u) ? ~u : (u | 0x80000000u);
  keys[t] = ((unsigned long long)(~u) << 32) | (unsigned)n;
  // shifted anchor
  float sx = (float)(w * 16), sy = (float)(h * 16);
  float ax1 = anchors[a*4+0] + sx;
  float ay1 = anchors[a*4+1] + sy;
  float ax2 = anchors[a*4+2] + sx;
  float ay2 = anchors[a*4+3] + sy;
  float aw  = ax2 - ax1 + 1.0f;
  float ah  = ay2 - ay1 + 1.0f;
  float acx = ax1 + 0.5f * aw;
  float acy = ay1 + 0.5f * ah;
  // deltas[b, a*4+k, h, w]
  size_t hwsz  = (size_t)HH_ * WW_;
  size_t dbase = (((size_t)b * (4*AA_)) * HH_ + h) * WW_ + w;
  float dx = deltas[dbase + (size_t)(a*4+0) * hwsz];
  float dy = deltas[dbase + (size_t)(a*4+1) * hwsz];
  float dw = deltas[dbase + (size_t)(a*4+2) * hwsz];
  float dh = deltas[dbase + (size_t)(a*4+3) * hwsz];
  float pcx = dx * aw + acx;
  float pcy = dy * ah + acy;
  float pw  = expf(dw) * aw;
  float ph  = expf(dh) * ah;
  float xmax = (float)(imgw[0] - 1);
  float ymax = (float)(imgh[0] - 1);
  float x1 = fminf(fmaxf(pcx - 0.5f * pw, 0.0f), xmax);
  float y1 = fminf(fmaxf(pcy - 0.5f * ph, 0.0f), ymax);
  float x2 = fminf(fmaxf(pcx + 0.5f * pw, 0.0f), xmax);
  float y2 = fminf(fmaxf(pcy + 0.5f * ph, 0.0f), ymax);
  size_t o = ((size_t)b * NN_ + n) * 4;
  proposals[o+0] = x1; proposals[o+1] = y1;
  proposals[o+2] = x2; proposals[o+3] = y2;
}

// ---------------- 2a) LDS-resident bitonic: all stages k=2..CHUNK_ in one launch ----------------
__global__ void __launch_bounds__(LTH_) sort_local_full(unsigned long long* __restrict__ keys) {
  __shared__ unsigned long long lk[CHUNK_];
  int tid = threadIdx.x;
  size_t base = (size_t)blockIdx.x * CHUNK_;
  for (int e = tid; e < CHUNK_; e += LTH_) lk[e] = keys[base + e];
  __syncthreads();
  int rbase = (int)(base & (NPAD_ - 1));   // position within batch segment
  for (int k = 2; k <= CHUNK_; k <<= 1) {
    for (int j = k >> 1; j >= 1; j >>= 1) {
#pragma unroll
      for (int q = tid; q < CHUNK_ / 2; q += LTH_) {
        int e = ((q & ~(j - 1)) << 1) | (q & (j - 1));
        int p = e | j;
        bool up = (((rbase + e) & k) == 0);
        unsigned long long x = lk[e], y = lk[p];
        if ((x > y) == up) { lk[e] = y; lk[p] = x; }
      }
      __syncthreads();
    }
  }
  for (int e = tid; e < CHUNK_; e += LTH_) keys[base + e] = lk[e];
}

// ---------------- 2b) LDS-resident tail: passes j=CHUNK_/2..1 of stage k (>=2*CHUNK_) ----------------
__global__ void __launch_bounds__(LTH_) sort_local_tail(unsigned long long* __restrict__ keys, int k) {
  __shared__ unsigned long long lk[CHUNK_];
  int tid = threadIdx.x;
  size_t base = (size_t)blockIdx.x * CHUNK_;
  for (int e = tid; e < CHUNK_; e += LTH_) lk[e] = keys[base + e];
  __syncthreads();
  int rbase = (int)(base & (NPAD_ - 1));
  bool up = ((rbase & k) == 0);            // uniform per chunk since k >= 2*CHUNK_
  for (int j = CHUNK_ / 2; j >= 1; j >>= 1) {
#pragma unroll
    for (int q = tid; q < CHUNK_ / 2; q += LTH_) {
      int e = ((q & ~(j - 1)) << 1) | (q & (j - 1));
      int p = e | j;
      unsigned long long x = lk[e], y = lk[p];
      if ((x > y) == up) { lk[e] = y; lk[p] = x; }
    }
    __syncthreads();
  }
  for (int e = tid; e < CHUNK_; e += LTH_) keys[base + e] = lk[e];
}

// ---------------- 2c) global bitonic pass (only j >= CHUNK_ cross-chunk passes) ----------------
__global__ void bitonic_pass(unsigned long long* __restrict__ keys, int k, int j) {
  int t = blockIdx.x * blockDim.x + threadIdx.x;
  if (t >= BB_ * NPAD_) return;
  int r = t & (NPAD_ - 1);
  int p = r ^ j;
  if (p > r) {
    size_t base = (size_t)(t >> 16) << 16;
    unsigned long long x = keys[base + r];
    unsigned long long y = keys[base + p];
    bool up = ((r & k) == 0);
    if ((x > y) == up) { keys[base + r] = y; keys[base + p] = x; }
  }
}

// ---------------- 3) gather top-6000 boxes (score order) + areas ----------------
__global__ void gather_kernel(const unsigned long long* __restrict__ keys,
                              const float* __restrict__ proposals,
                              float* __restrict__ boxes,
                              float* __restrict__ areas) {
  int t = blockIdx.x * blockDim.x + threadIdx.x;
  if (t >= BB_ * PRE_) return;
  int b = t / PRE_;
  unsigned long long k = keys[((size_t)b << 16) + (t % PRE_)];
  unsigned n = (unsigned)(k & 0xFFFFFFFFu);
  size_t src = ((size_t)b * NN_ + n) * 4;
  float x1 = proposals[src+0], y1 = proposals[src+1];
  float x2 = proposals[src+2], y2 = proposals[src+3];
  size_t o = (size_t)t * 4;
  boxes[o+0] = x1; boxes[o+1] = y1; boxes[o+2] = x2; boxes[o+3] = y2;
  areas[t] = (x2 - x1 + 1.0f) * (y2 - y1 + 1.0f);
}

// ---------------- 4) suppression mask: one wave per 16x16 tile, WMMA denominators ----------------
__global__ void __launch_bounds__(32) nms_mask_kernel(
    const float* __restrict__ boxes,       // [B][6000][4]
    const float* __restrict__ areas,       // [B][6000]
    unsigned short* __restrict__ mask) {   // [B][6000][375] 16 cols per word
  int it = blockIdx.x, jt = blockIdx.y, b = blockIdx.z;
  int lane = threadIdx.x;
  int i0 = it * 16, j0 = jt * 16;
  __shared__ float sBI[16][4], sBJ[16][4], sAI[16], sAJ[16];
  const float* bb = boxes + (size_t)b * PRE_ * 4;
  const float* aa = areas + (size_t)b * PRE_;
  if (lane < 16) {
    int i = i0 + lane;
    sBI[lane][0] = bb[i*4+0]; sBI[lane][1] = bb[i*4+1];
    sBI[lane][2] = bb[i*4+2]; sBI[lane][3] = bb[i*4+3];
    sAI[lane] = aa[i];
  } else {
    int l = lane - 16, j = j0 + l;
    sBJ[l][0] = bb[j*4+0]; sBJ[l][1] = bb[j*4+1];
    sBJ[l][2] = bb[j*4+2]; sBJ[l][3] = bb[j*4+3];
    sAJ[l] = aa[j];
  }
  __syncthreads();
  // D[i][j] = area_i*1 + 1*area_j via rank-2 WMMA (K=2,3 zero-padded).
  // A 16x4 f32 layout: lanes 0-15 hold K=0,1 for M=lane; lanes 16-31 hold K=2,3.
  v2f av; av.x = 0.0f; av.y = 0.0f;
  v2f bv; bv.x = 0.0f; bv.y = 0.0f;
  if (lane < 16) {
    av.x = sAI[lane]; av.y = 1.0f;        // row i: [area_i, 1, 0, 0]
    bv.x = 1.0f;      bv.y = sAJ[lane];   // col j: [1, area_j, 0, 0]
  }
  v8f c = {};
  v8f d = __builtin_amdgcn_wmma_f32_16x16x4_f32(
      /*neg_a=*/false, av, /*neg_b=*/false, bv,
      /*c_mod=*/(short)0, c, /*reuse_a=*/false, /*reuse_b=*/false);
  // C/D layout: VGPR v -> row v (lanes 0-15) / row v+8 (lanes 16-31); lane&15 -> col
  int half = lane >> 4;
  int col  = lane & 15;
  int j    = j0 + col;
  float jx1 = sBJ[col][0], jy1 = sBJ[col][1], jx2 = sBJ[col][2], jy2 = sBJ[col][3];
#pragma unroll
  for (int v = 0; v < 8; ++v) {
    int row = v + half * 8;
    int i = i0 + row;
    float ix1 = sBI[row][0], iy1 = sBI[row][1], ix2 = sBI[row][2], iy2 = sBI[row][3];
    float iw = fminf(ix2, jx2) - fmaxf(ix1, jx1) + 1.0f;
    float ih = fminf(iy2, jy2) - fmaxf(iy1, jy1) + 1.0f;
    iw = fmaxf(iw, 0.0f); ih = fmaxf(ih, 0.0f);
    float inter = iw * ih;
    float denom = d[v] - inter;                 // (area_i + area_j) from WMMA
    bool sup = (inter > THRESH_ * denom) && (j > i);
    unsigned bal = (unsigned)__ballot(sup);     // wave32: low16=row v, high16=row v+8
    if (lane == 0) {
      size_t rb = (size_t)b * PRE_;
      mask[(rb + (size_t)(i0 + v))     * NT_ + jt] = (unsigned short)(bal & 0xFFFFu);
      mask[(rb + (size_t)(i0 + v + 8)) * NT_ + jt] = (unsigned short)(bal >> 16);
    }
  }
}

// ---------------- 5) per-batch greedy reduce, emit first 300 kept ----------------
__global__ void __launch_bounds__(128) nms_reduce_kernel(
    const float* __restrict__ boxes,
    const unsigned short* __restrict__ mask,
    float* __restrict__ out) {
  int b = blockIdx.x;
  int tid = threadIdx.x;
  __shared__ unsigned short removed[NT_ + 1];
  for (int t = tid; t < NT_ + 1; t += blockDim.x) removed[t] = 0;
  __syncthreads();
  const float* bb = boxes + (size_t)b * PRE_ * 4;
  float bf = (float)b;
  int cnt = 0;
  for (int i = 0; i < PRE_; ++i) {
    bool kept = (((unsigned)removed[i >> 4] >> (i & 15)) & 1u) == 0u;
    if (kept) {
      if (tid == 0) {
        size_t o = ((size_t)b * POST_ + cnt) * 5;
        out[o+0] = bf;
        out[o+1] = bb[i*4+0]; out[o+2] = bb[i*4+1];
        out[o+3] = bb[i*4+2]; out[o+4] = bb[i*4+3];
      }
      cnt++;
      if (cnt >= POST_) break;
      const unsigned short* mrow = mask + ((size_t)b * PRE_ + i) * NT_;
      if (i + 1 < PRE_)
        __builtin_prefetch(mrow + NT_, 0, 1);   // global_prefetch_b8 next row
      for (int t = tid; t < NT_; t += blockDim.x) removed[t] |= mrow[t];
      __syncthreads();
    }
  }
  // zero-fill remaining ranks (batch index still present, boxes = 0)
  for (int r = cnt + tid; r < POST_; r += blockDim.x) {
    size_t o = ((size_t)b * POST_ + r) * 5;
    out[o+0] = bf; out[o+1] = 0.0f; out[o+2] = 0.0f; out[o+3] = 0.0f; out[o+4] = 0.0f;
  }
}

extern "C" void kernel_launch(void* const* d_in, const int* in_sizes, int n_in,
                              void* d_out, int out_size, void* d_ws, size_t ws_size,
                              hipStream_t stream) {
  const float* scores  = (const float*)d_in[0];
  const float* deltas  = (const float*)d_in[1];
  const float* anchors = (const float*)d_in[2];
  const int*   imgw    = (const int*)d_in[3];
  const int*   imgh    = (const int*)d_in[4];
  float* out = (float*)d_out;
  char* ws = (char*)d_ws;
  unsigned long long* keys = (unsigned long long*)(ws + OFF_KEYS);
  float* proposals = (float*)(ws + OFF_PROP);
  float* boxes     = (float*)(ws + OFF_BOX);
  float* areas     = (float*)(ws + OFF_AREA);
  unsigned short* mask = (unsigned short*)(ws + OFF_MASK);

  {
    int total = BB_ * NPAD_;
    decode_kernel<<<(total + 255) / 256, 256, 0, stream>>>(
        scores, deltas, anchors, imgw, imgh, proposals, keys);
  }
  {
    int nchunks = BB_ * NPAD_ / CHUNK_;                  // 128 blocks
    int gblocks = (BB_ * NPAD_ + 255) / 256;
    // stages k = 2..CHUNK_ entirely in LDS (one launch)
    sort_local_full<<<nchunks, LTH_, 0, stream>>>(keys);
    // stages k = 2*CHUNK_..NPAD_: cross-chunk passes global, tail passes in LDS
    for (int k = CHUNK_ * 2; k <= NPAD_; k <<= 1) {
      for (int j = k >> 1; j >= CHUNK_; j >>= 1)
        bitonic_pass<<<gblocks, 256, 0, stream>>>(keys, k, j);
      sort_local_tail<<<nchunks, LTH_, 0, stream>>>(keys, k);
    }
  }
  {
    int total = BB_ * PRE_;
    gather_kernel<<<(total + 255) / 256, 256, 0, stream>>>(keys, proposals, boxes, areas);
  }
  {
    dim3 grid(NT_, NT_, BB_);
    nms_mask_kernel<<<grid, 32, 0, stream>>>(boxes, areas, mask);
  }
  nms_reduce_kernel<<<BB_, 128, 0, stream>>>(boxes, mask, out);
}